// vsl_gg_crf_88278757802456
// MI455X (gfx1250) — compile-verified
//
#include <hip/hip_runtime.h>

typedef __attribute__((ext_vector_type(2))) float v2f;
typedef __attribute__((ext_vector_type(8))) float v8f;

#define Bsz 64
#define Tsz 256
#define Hsz 200
#define Ksz 48
#define MROWS (Bsz * Tsz)      // 16384 rows of the emission matrix
#define TILE_STRIDE 49         // 48 cols + 1 pad -> conflict-free row reads

// ---------------------------------------------------------------------------
// Kernel 1: emission GEMM (f32 WMMA 16x16x4) + per-(b,t) LSE / gold / end_tag
// grid: 128 blocks x 256 threads = 1024 waves; one 16-row M-tile per wave.
// ---------------------------------------------------------------------------
__global__ __launch_bounds__(256)
void crf_emission_lse_kernel(const float* __restrict__ feats,   // [B*T, H]
                             const float* __restrict__ W_e,     // [K, H]
                             const float* __restrict__ b_e,     // [K]
                             const int*   __restrict__ targets, // [B*T] (stride 1 or 2)
                             int tgt_stride,
                             const int*   __restrict__ end_tag_p,
                             float* __restrict__ lse_out,       // [B*T]
                             float* __restrict__ gold_out,      // [B*T]
                             float* __restrict__ emend_out)     // [B*T]
{
    __shared__ float sW[Ksz * Hsz];                 // 38400 B: W_e staged in LDS
    __shared__ float sTile[8][16 * TILE_STRIDE];    // per-wave 16x48 emission tile

    const int tid  = threadIdx.x;
    const int lane = tid & 31;
    const int wave = tid >> 5;
    const int half = lane >> 4;    // 0: lanes 0-15, 1: lanes 16-31
    const int l16  = lane & 15;

    // Stage W_e into LDS (reused by all 8 waves, all K-chunks).
    for (int i = tid; i < Ksz * Hsz; i += 256) sW[i] = W_e[i];
    __syncthreads();

    const int tile = blockIdx.x * 8 + wave;     // 0..1023
    const int row  = tile * 16 + l16;           // this lane's A-matrix row (M = l16)

    // f32 16x16x4 WMMA fragments:
    //  A (16x4, MxK): lane M = lane%16; K = 2*(lane/16) + vgpr   -> float2 load
    //  B (4x16, KxN): lane N = lane%16; K = 2*(lane/16) + vgpr   -> float2 load
    //  B[k][n] = W_e[n][k]  (emission = feats . W_e^T)
    v8f c0 = {}; v8f c1 = {}; v8f c2 = {};

    const float* aBase = feats + (size_t)row * Hsz + half * 2;
    const float* bBase0 = &sW[( 0 + l16) * Hsz + half * 2];
    const float* bBase1 = &sW[(16 + l16) * Hsz + half * 2];
    const float* bBase2 = &sW[(32 + l16) * Hsz + half * 2];

    #pragma unroll 2
    for (int k0 = 0; k0 < Hsz; k0 += 4) {
        v2f a   = *(const v2f*)(aBase  + k0);
        v2f bb0 = *(const v2f*)(bBase0 + k0);
        v2f bb1 = *(const v2f*)(bBase1 + k0);
        v2f bb2 = *(const v2f*)(bBase2 + k0);
        c0 = __builtin_amdgcn_wmma_f32_16x16x4_f32(false, a, false, bb0, (short)0, c0, false, false);
        c1 = __builtin_amdgcn_wmma_f32_16x16x4_f32(false, a, false, bb1, (short)0, c1, false, false);
        c2 = __builtin_amdgcn_wmma_f32_16x16x4_f32(false, a, false, bb2, (short)0, c2, false, false);
    }

    // Spill accumulators (+ bias) to this wave's LDS tile.
    // C/D layout: lane N = lane%16; VGPR i holds M = i + 8*(lane/16).
    const float bias0 = b_e[ 0 + l16];
    const float bias1 = b_e[16 + l16];
    const float bias2 = b_e[32 + l16];
    float* tp = sTile[wave];
    #pragma unroll
    for (int i = 0; i < 8; ++i) {
        const int r = i + half * 8;
        tp[r * TILE_STRIDE +  0 + l16] = c0[i] + bias0;
        tp[r * TILE_STRIDE + 16 + l16] = c1[i] + bias1;
        tp[r * TILE_STRIDE + 32 + l16] = c2[i] + bias2;
    }
    __syncthreads();   // uniform; orders LDS writes before per-row reads

    // Lanes 0..15: one emission row each -> LSE over 48, gold gather, end_tag.
    if (half == 0) {
        const float* rp = &sTile[wave][l16 * TILE_STRIDE];
        float m = rp[0];
        #pragma unroll
        for (int k = 1; k < Ksz; ++k) m = fmaxf(m, rp[k]);
        float s = 0.0f;
        #pragma unroll
        for (int k = 0; k < Ksz; ++k) s += expf(rp[k] - m);
        const float lse = m + logf(s);

        const int g   = tile * 16 + l16;
        int tgt = targets[(size_t)g * tgt_stride] % Ksz;
        if (tgt < 0) tgt += Ksz;
        lse_out[g]   = lse;
        gold_out[g]  = rp[tgt];
        emend_out[g] = rp[*end_tag_p];
    }
}

// ---------------------------------------------------------------------------
// Kernel 2: per-batch reduction over T.
// loss[b] = emend[b, L-1] + sum_{t<L-1} lse[b,t] - sum_{t<L} gold[b,t]
// ---------------------------------------------------------------------------
__global__ __launch_bounds__(256)
void crf_reduce_kernel(const float* __restrict__ lse,
                       const float* __restrict__ gold,
                       const float* __restrict__ emend,
                       const int*   __restrict__ mask,
                       float* __restrict__ out)
{
    __shared__ float red[Tsz];
    __shared__ int   ired[Tsz];

    const int b = blockIdx.x;
    const int t = threadIdx.x;

    ired[t] = mask[b * Tsz + t];
    __syncthreads();
    for (int s = Tsz / 2; s > 0; s >>= 1) {
        if (t < s) ired[t] += ired[t + s];
        __syncthreads();
    }
    const int L = ired[0];

    float contrib = 0.0f;
    if (t < L)     contrib -= gold[b * Tsz + t];
    if (t < L - 1) contrib += lse[b * Tsz + t];
    red[t] = contrib;
    __syncthreads();
    for (int s = Tsz / 2; s > 0; s >>= 1) {
        if (t < s) red[t] += red[t + s];
        __syncthreads();
    }
    if (t == 0) {
        const float all_paths_tail = (L > 0) ? emend[b * Tsz + (L - 1)] : 0.0f;
        out[b] = red[0] + all_paths_tail;
    }
}

// ---------------------------------------------------------------------------
extern "C" void kernel_launch(void* const* d_in, const int* in_sizes, int n_in,
                              void* d_out, int out_size, void* d_ws, size_t ws_size,
                              hipStream_t stream) {
    const float* feats   = (const float*)d_in[0];  // [B,T,H] f32
    const float* W_e     = (const float*)d_in[1];  // [K,H]   f32
    const float* b_e     = (const float*)d_in[2];  // [K]     f32
    const int*   targets = (const int*)d_in[3];    // [B,T]   int (maybe i64 pairs)
    const int*   mask    = (const int*)d_in[4];    // [B,T]   int32
    const int*   end_tag = (const int*)d_in[6];    // scalar
    float* out = (float*)d_out;                    // [B] f32

    // targets declared int64 in the reference; detect layout from flat size.
    const int tgt_stride = (in_sizes[3] >= 2 * MROWS) ? 2 : 1;

    float* lse   = (float*)d_ws;          // [B*T]
    float* goldv = lse + MROWS;           // [B*T]
    float* emend = goldv + MROWS;         // [B*T]  (total 192 KB of d_ws)

    crf_emission_lse_kernel<<<MROWS / (16 * 8), 256, 0, stream>>>(
        feats, W_e, b_e, targets, tgt_stride, end_tag, lse, goldv, emend);

    crf_reduce_kernel<<<Bsz, Tsz, 0, stream>>>(lse, goldv, emend, mask, out);
}